// FCGFAutoencoder_36283883717338
// MI455X (gfx1250) — compile-verified
//
#include <hip/hip_runtime.h>
#include <hip/hip_bf16.h>

// ---------------------------------------------------------------------------
// Problem constants
// ---------------------------------------------------------------------------
#define NPTS   4194304   // N points
#define NCH    32        // channels
#define NB     64        // batches
#define NPOUT  3072      // 3 * 1024 output per batch

// segment-max launch geometry
#define SEG_BLOCKS  1024
#define SEG_THREADS 256
#define SEG_WAVES   (SEG_BLOCKS * (SEG_THREADS / 32))       // 8192
#define PTS_PER_WAVE (NPTS / SEG_WAVES)                      // 512 (mult of 4)

typedef float v2f __attribute__((ext_vector_type(2)));
typedef float v8f __attribute__((ext_vector_type(8)));

// ---------------------------------------------------------------------------
// Order-preserving float <-> u32 key encoding (monotonic for all non-NaN).
// key 0 == "never written" sentinel (decodes to non-finite -> 0.0 like the
// reference's where(isfinite(g), g, 0)).
// ---------------------------------------------------------------------------
__device__ __forceinline__ unsigned enc_f32(float f) {
  unsigned u = __float_as_uint(f);
  return (u & 0x80000000u) ? ~u : (u | 0x80000000u);
}
__device__ __forceinline__ float dec_key(unsigned k) {
  unsigned u = (k & 0x80000000u) ? (k ^ 0x80000000u) : ~k;
  // non-finite (incl. sentinel 0 and -inf) -> 0.0f
  if ((u & 0x7F800000u) == 0x7F800000u) return 0.0f;
  return __uint_as_float(u);
}

// ---------------------------------------------------------------------------
// Zero the global key buffer (64*32 u32)
// ---------------------------------------------------------------------------
__global__ void fcgf_init_keys(unsigned* __restrict__ gkeys) {
  int i = blockIdx.x * blockDim.x + threadIdx.x;
  if (i < NB * NCH) gkeys[i] = 0u;
}

// ---------------------------------------------------------------------------
// Streaming segment max: features [N,32] f32, sorted batch_ids [N] i32
// -> gkeys [64,32] encoded u32.
// Each wave owns PTS_PER_WAVE contiguous points. Lane layout: lane = 8*subp
// groupings -- lane handles point (p + (lane>>3)) channels (lane&7)*4..+3 via
// one float4 load (wave-coalesced 512B). Running max kept in registers,
// flushed to LDS atomics only on (rare) batch-id changes, then block-combined
// into global atomics.
// ---------------------------------------------------------------------------
__device__ __forceinline__ void lds_max4(unsigned* lk, int b, int cg, float4 m) {
  atomicMax(&lk[b * NCH + cg + 0], enc_f32(m.x));
  atomicMax(&lk[b * NCH + cg + 1], enc_f32(m.y));
  atomicMax(&lk[b * NCH + cg + 2], enc_f32(m.z));
  atomicMax(&lk[b * NCH + cg + 3], enc_f32(m.w));
}

__global__ void fcgf_segment_max(const float* __restrict__ feat,
                                 const int* __restrict__ ids,
                                 unsigned* __restrict__ gkeys) {
  __shared__ unsigned lkeys[NB * NCH];
  for (int i = threadIdx.x; i < NB * NCH; i += blockDim.x) lkeys[i] = 0u;
  __syncthreads();

  const int lane = threadIdx.x & 31;
  const int wave = (blockIdx.x * blockDim.x + threadIdx.x) >> 5;
  const int p0   = wave * PTS_PER_WAVE;
  const int subp = lane >> 3;        // which of 4 points in the group
  const int cg   = (lane & 7) * 4;   // channel group base (0,4,...,28)

  const float NEG = -__builtin_huge_valf();
  float4 m = make_float4(NEG, NEG, NEG, NEG);
  int cur = ids[p0];

#pragma unroll 4
  for (int p = p0; p < p0 + PTS_PER_WAVE; p += 4) {
    const int bLast = ids[p + 3];  // uniform across wave
    const float4 v = *reinterpret_cast<const float4*>(
        feat + ((size_t)(p + subp) << 5) + cg);
    if (bLast == cur) {            // fast path: whole group in current batch
      m.x = fmaxf(m.x, v.x); m.y = fmaxf(m.y, v.y);
      m.z = fmaxf(m.z, v.z); m.w = fmaxf(m.w, v.w);
    } else {                       // boundary: flush + direct per-point merge
      lds_max4(lkeys, cur, cg, m);
      m = make_float4(NEG, NEG, NEG, NEG);
      const int b = ids[p + subp]; // per-lane batch of this group's point
      lds_max4(lkeys, b, cg, v);
      cur = bLast;
    }
  }
  lds_max4(lkeys, cur, cg, m);

  __syncthreads();
  for (int i = threadIdx.x; i < NB * NCH; i += blockDim.x) {
    unsigned k = lkeys[i];
    if (k) atomicMax(&gkeys[i], k);
  }
}

// ---------------------------------------------------------------------------
// WMMA f32 GEMM: Out[64,Nn] = act(A[64,K] @ W[K,Nn] + bias)
// One wave (32 threads) per 16x16 output tile, K-loop of V_WMMA_F32_16X16X4_F32.
// ISA 7.12.2 layouts:
//   A 16x4 f32 : lane m = l&15, half = l>>4 ; reg0 = A[m][k+2*half], reg1 = +1
//   B 4x16 f32 : lane n = l&15, half       ; reg0 = B[k+2*half][n],  reg1 = +1
//   C/D 16x16  : VGPR r -> row r + 8*half, col = l&15
// DECODE: A is the encoded segment-max key buffer (layer 1 input).
// ---------------------------------------------------------------------------
template <bool RELU, bool DECODE>
__global__ void fcgf_mlp_gemm(const void* __restrict__ Ain,
                              const float* __restrict__ W,
                              const float* __restrict__ bias,
                              float* __restrict__ Out,
                              int K, int Nn) {
  const int lane = threadIdx.x;   // 0..31, full wave => EXEC all ones
  const int nt   = blockIdx.x;    // N tile
  const int mt   = blockIdx.y;    // M tile (0..3)
  const int half = lane >> 4;
  const int mn   = lane & 15;     // row for A, col for B/D
  const int k0   = half * 2;

  const float*    Af = (const float*)Ain;
  const unsigned* Au = (const unsigned*)Ain;
  const size_t arow = (size_t)(mt * 16 + mn) * (size_t)K;
  const int    ncol = nt * 16 + mn;

  v8f acc = {};
  for (int k = 0; k < K; k += 4) {
    v2f a, b;
    if (DECODE) {
      a.x = dec_key(Au[arow + k + k0]);
      a.y = dec_key(Au[arow + k + k0 + 1]);
    } else {
      a.x = Af[arow + k + k0];
      a.y = Af[arow + k + k0 + 1];
    }
    b.x = W[(size_t)(k + k0) * Nn + ncol];
    b.y = W[(size_t)(k + k0 + 1) * Nn + ncol];
    acc = __builtin_amdgcn_wmma_f32_16x16x4_f32(
        /*neg_a=*/false, a, /*neg_b=*/false, b,
        /*c_mod=*/(short)0, acc, /*reuse_a=*/false, /*reuse_b=*/false);
  }

  const float bv = bias[ncol];
#pragma unroll
  for (int r = 0; r < 8; ++r) {
    float v = acc[r] + bv;
    if (RELU) v = fmaxf(v, 0.0f);
    Out[(size_t)(mt * 16 + r + 8 * half) * Nn + ncol] = v;
  }
}

// ---------------------------------------------------------------------------
// Launch
// ---------------------------------------------------------------------------
extern "C" void kernel_launch(void* const* d_in, const int* in_sizes, int n_in,
                              void* d_out, int out_size, void* d_ws, size_t ws_size,
                              hipStream_t stream) {
  (void)in_sizes; (void)n_in; (void)out_size; (void)ws_size;

  const float* feat = (const float*)d_in[0];
  const int*   ids  = (const int*)d_in[1];
  const float* W1   = (const float*)d_in[2];
  const float* b1   = (const float*)d_in[3];
  const float* W2   = (const float*)d_in[4];
  const float* b2   = (const float*)d_in[5];
  const float* W3   = (const float*)d_in[6];
  const float* b3   = (const float*)d_in[7];
  float* out = (float*)d_out;

  unsigned char* ws = (unsigned char*)d_ws;
  unsigned* gkeys = (unsigned*)ws;                              // 64*32*4   = 8 KB
  float*    h1    = (float*)(ws + 8 * 1024);                    // 64*256*4  = 64 KB
  float*    h2    = (float*)(ws + 8 * 1024 + 64 * 1024);        // 64*512*4  = 128 KB

  // 1) init segment-max key buffer
  fcgf_init_keys<<<(NB * NCH + 255) / 256, 256, 0, stream>>>(gkeys);

  // 2) bandwidth-bound streaming segment max (512 MB @ HBM rate)
  fcgf_segment_max<<<SEG_BLOCKS, SEG_THREADS, 0, stream>>>(feat, ids, gkeys);

  // 3) MLP via f32 WMMA (M=64 -> 4 row tiles)
  fcgf_mlp_gemm<true, true ><<<dim3(256  / 16, 4), 32, 0, stream>>>(gkeys, W1, b1, h1, NCH, 256);
  fcgf_mlp_gemm<true, false><<<dim3(512  / 16, 4), 32, 0, stream>>>(h1,    W2, b2, h2, 256, 512);
  fcgf_mlp_gemm<false,false><<<dim3(NPOUT/ 16, 4), 32, 0, stream>>>(h2,    W3, b3, out, 512, NPOUT);
}